// Corener_57062935495475
// MI455X (gfx1250) — compile-verified
//
#include <hip/hip_runtime.h>

#define Bc 2
#define Sc 512
#define Hc 768
#define Ec 64
#define Mc 64
#define Rc 128
#define SIZE_E 32
#define CLSTOK 101
#define NEGV (-1e30f)

typedef __attribute__((ext_vector_type(2))) float v2f;
typedef __attribute__((ext_vector_type(8))) float v8f;

// ---------------------------------------------------------------------------
// CLS-token context: ctx[b,:] = emb[b, argmax(ids==CLS), :]
// ---------------------------------------------------------------------------
__global__ void ctx_kernel(const int* __restrict__ ids,
                           const float* __restrict__ emb,
                           float* __restrict__ ctx) {
  __shared__ int sidx;
  int b = blockIdx.x;
  if (threadIdx.x == 0) sidx = Sc;  // sentinel
  __syncthreads();
  for (int s = threadIdx.x; s < Sc; s += blockDim.x)
    if (ids[b * Sc + s] == CLSTOK) atomicMin(&sidx, s);
  __syncthreads();
  int ci = (sidx < Sc) ? sidx : 0;  // argmax of all-zero mask -> 0
  for (int h = threadIdx.x; h < Hc; h += blockDim.x)
    ctx[b * Hc + h] = emb[((size_t)b * Sc + ci) * Hc + h];
}

// ---------------------------------------------------------------------------
// Size-embedding gather: out[b,k,:] = table[sizes[b,k], :]
// ---------------------------------------------------------------------------
__global__ void sz_kernel(const int* __restrict__ sizes,
                          const float* __restrict__ table,
                          float* __restrict__ out, int total) {
  int idx = blockIdx.x * blockDim.x + threadIdx.x;
  if (idx >= total) return;
  int e = idx / SIZE_E, c = idx % SIZE_E;
  out[idx] = table[sizes[e] * SIZE_E + c];
}

// ---------------------------------------------------------------------------
// Masked max pooling: out[b,k,h] = max_s( emb[b,s,h] + (mask[b,k,s]?0:NEG) )
// GUARD: zero the row when no mask bit is set (relation-context variant).
// TK=8 spans per block; masks staged in LDS; one emb pass per 8 spans.
// ---------------------------------------------------------------------------
template <bool GUARD>
__global__ __launch_bounds__(256) void pool_kernel(
    const int* __restrict__ masks,   // [B,K,S]
    const float* __restrict__ emb,   // [B,S,H]
    float* __restrict__ out,         // [B,K,H]
    int Ktot) {
  constexpr int TK = 8;
  __shared__ int smask[TK * Sc];
  __shared__ unsigned sany;
  int groups = Ktot / TK;
  int b = blockIdx.x / groups;
  int k0 = (blockIdx.x % groups) * TK;

  if (threadIdx.x == 0) sany = 0u;
  __syncthreads();
  const int* mptr = masks + ((size_t)b * Ktot + k0) * Sc;
  unsigned local = 0u;
  for (int i = threadIdx.x; i < TK * Sc; i += 256) {
    int v = mptr[i];
    smask[i] = v;
    if (v) local |= 1u << (i / Sc);
  }
  if (GUARD && local) atomicOr(&sany, local);
  __syncthreads();

  float acc[TK][3];
#pragma unroll
  for (int t = 0; t < TK; t++)
#pragma unroll
    for (int j = 0; j < 3; j++) acc[t][j] = -INFINITY;

  for (int s = 0; s < Sc; s++) {
    const float* ep = emb + ((size_t)b * Sc + s) * Hc;
    float e0 = ep[threadIdx.x];
    float e1 = ep[threadIdx.x + 256];
    float e2 = ep[threadIdx.x + 512];
#pragma unroll
    for (int t = 0; t < TK; t++) {
      float add = smask[t * Sc + s] ? 0.0f : NEGV;
      acc[t][0] = fmaxf(acc[t][0], e0 + add);
      acc[t][1] = fmaxf(acc[t][1], e1 + add);
      acc[t][2] = fmaxf(acc[t][2], e2 + add);
    }
  }

  unsigned anyb = GUARD ? sany : 0xFFFFFFFFu;
#pragma unroll
  for (int t = 0; t < TK; t++) {
    bool zero = GUARD && !((anyb >> t) & 1u);
    float* op = out + ((size_t)b * Ktot + k0 + t) * Hc;
#pragma unroll
    for (int j = 0; j < 3; j++)
      op[threadIdx.x + 256 * j] = zero ? 0.0f : acc[t][j];
  }
}

// ---------------------------------------------------------------------------
// Span rep: rep[b,k,:] = [ ctx(b) | pool(b,k) | sz(b,k) ]  (width 2H+SIZE_E)
// ---------------------------------------------------------------------------
__global__ void span_rep_kernel(const float* __restrict__ ctx,
                                const float* __restrict__ pool,
                                const float* __restrict__ sz,
                                float* __restrict__ rep, int K, int total) {
  const int W = 2 * Hc + SIZE_E;
  int idx = blockIdx.x * blockDim.x + threadIdx.x;
  if (idx >= total) return;
  int bk = idx / W, c = idx % W;
  int b = bk / K;
  float v;
  if (c < Hc)            v = ctx[b * Hc + c];
  else if (c < 2 * Hc)   v = pool[(size_t)bk * Hc + (c - Hc)];
  else                   v = sz[(size_t)bk * SIZE_E + (c - 2 * Hc)];
  rep[idx] = v;
}

// ---------------------------------------------------------------------------
// Relation rep: rep[b,r,:] = [ rctx | pool[i0] | pool[i1] | sz[i0] | sz[i1] ]
// width 3H + 2*SIZE_E
// ---------------------------------------------------------------------------
__global__ void rel_rep_kernel(const float* __restrict__ rctx,
                               const float* __restrict__ pool,
                               const float* __restrict__ sz,
                               const int* __restrict__ rels,
                               float* __restrict__ rep, int R, int Kpool,
                               int total) {
  const int W = 3 * Hc + 2 * SIZE_E;
  int idx = blockIdx.x * blockDim.x + threadIdx.x;
  if (idx >= total) return;
  int br = idx / W, c = idx % W;
  int b = br / R;
  int i0 = rels[br * 2 + 0], i1 = rels[br * 2 + 1];
  float v;
  if (c < Hc)                  v = rctx[(size_t)br * Hc + c];
  else if (c < 2 * Hc)         v = pool[((size_t)b * Kpool + i0) * Hc + (c - Hc)];
  else if (c < 3 * Hc)         v = pool[((size_t)b * Kpool + i1) * Hc + (c - 2 * Hc)];
  else if (c < 3 * Hc + SIZE_E) v = sz[((size_t)b * Kpool + i0) * SIZE_E + (c - 3 * Hc)];
  else                          v = sz[((size_t)b * Kpool + i1) * SIZE_E + (c - 3 * Hc - SIZE_E)];
  rep[idx] = v;
}

// ---------------------------------------------------------------------------
// WMMA GEMM (f32, 16x16x4): C[M,N] = relu(A[M,K] @ W[N,K]^T + bias[N])
// One wave per 16x16 C tile; 8 waves per block. M,N mult of 16, K mult of 4.
// A frag:   lane<16 -> {A[m][k],A[m][k+1]}, lane>=16 -> {A[m][k+2],A[m][k+3]}
// B=W^T frag has the symmetric layout -> identical float2 addressing on W.
// ---------------------------------------------------------------------------
__global__ __launch_bounds__(256) void wmma_gemm_relu(
    const float* __restrict__ A, const float* __restrict__ W,
    const float* __restrict__ bias, float* __restrict__ C, int Mr, int Nc,
    int Kd) {
  int wave = threadIdx.x >> 5;
  int lane = threadIdx.x & 31;
  int tilesN = Nc >> 4;
  int tile = blockIdx.x * 8 + wave;
  int tm = tile / tilesN;
  int tn = tile - tm * tilesN;
  if (tm * 16 >= Mr) return;  // wave-uniform: EXEC stays all-ones for WMMA

  int hl = lane & 15;
  int koff = (lane < 16) ? 0 : 2;
  const float* arow = A + (size_t)(tm * 16 + hl) * Kd + koff;
  const float* wrow = W + (size_t)(tn * 16 + hl) * Kd + koff;

  v8f acc = {};
#pragma unroll 4
  for (int k = 0; k < Kd; k += 4) {
    v2f a = *(const v2f*)(arow + k);
    v2f b = *(const v2f*)(wrow + k);
    acc = __builtin_amdgcn_wmma_f32_16x16x4_f32(
        /*neg_a=*/false, a, /*neg_b=*/false, b,
        /*c_mod=*/(short)0, acc, /*reuse_a=*/false, /*reuse_b=*/false);
  }

  int n = tn * 16 + hl;
  float bv = bias[n];
  int rbase = tm * 16 + ((lane < 16) ? 0 : 8);
#pragma unroll
  for (int i = 0; i < 8; i++) {
    float v = acc[i] + bv;
    v = fmaxf(v, 0.0f);
    C[(size_t)(rbase + i) * Nc + n] = v;
  }
}

// ---------------------------------------------------------------------------
// Small output head: out[M,N] = A[M,K] @ W[N,K]^T + bias (N in {1,2,8,10})
// ---------------------------------------------------------------------------
__global__ void head_kernel(const float* __restrict__ A,
                            const float* __restrict__ W,
                            const float* __restrict__ bias,
                            float* __restrict__ out, int Mr, int Nc, int Kd) {
  int idx = blockIdx.x * blockDim.x + threadIdx.x;
  if (idx >= Mr * Nc) return;
  int m = idx / Nc, n = idx % Nc;
  const float* a = A + (size_t)m * Kd;
  const float* w = W + (size_t)n * Kd;
  float s = bias[n];
  for (int k = 0; k < Kd; k++) s = fmaf(a[k], w[k], s);
  out[idx] = s;
}

// ---------------------------------------------------------------------------
extern "C" void kernel_launch(void* const* d_in, const int* in_sizes, int n_in,
                              void* d_out, int out_size, void* d_ws,
                              size_t ws_size, hipStream_t stream) {
  const int* input_ids   = (const int*)d_in[0];
  const int* ent_masks   = (const int*)d_in[1];
  const int* ent_sizes   = (const int*)d_in[2];
  const int* men_masks   = (const int*)d_in[3];
  const int* men_sizes   = (const int*)d_in[4];
  const int* relations   = (const int*)d_in[5];
  const int* rel_masks   = (const int*)d_in[6];
  const int* references  = (const int*)d_in[7];
  const int* ref_masks   = (const int*)d_in[8];
  const float* emb       = (const float*)d_in[9];
  const float* ner_tab   = (const float*)d_in[10];
  const float* emd_tab   = (const float*)d_in[11];
  const float* ner_rep_w = (const float*)d_in[12];
  const float* ner_rep_b = (const float*)d_in[13];
  const float* ner_hd_w  = (const float*)d_in[14];
  const float* ner_hd_b  = (const float*)d_in[15];
  const float* emd_rep_w = (const float*)d_in[16];
  const float* emd_rep_b = (const float*)d_in[17];
  const float* emd_hd_w  = (const float*)d_in[18];
  const float* emd_hd_b  = (const float*)d_in[19];
  const float* rel_w1 = (const float*)d_in[20]; const float* rel_b1 = (const float*)d_in[21];
  const float* rel_w2 = (const float*)d_in[22]; const float* rel_b2 = (const float*)d_in[23];
  const float* rel_w3 = (const float*)d_in[24]; const float* rel_b3 = (const float*)d_in[25];
  const float* cr_w1  = (const float*)d_in[26]; const float* cr_b1  = (const float*)d_in[27];
  const float* cr_w2  = (const float*)d_in[28]; const float* cr_b2  = (const float*)d_in[29];
  const float* cr_w3  = (const float*)d_in[30]; const float* cr_b3  = (const float*)d_in[31];

  float* out = (float*)d_out;
  float* ws = (float*)d_ws;

  const int WS = 2 * Hc + SIZE_E;       // 1568
  const int WR = 3 * Hc + 2 * SIZE_E;   // 2368

  size_t o = 0;
  float* ctx   = ws + o; o += (size_t)Bc * Hc;
  float* epool = ws + o; o += (size_t)Bc * Ec * Hc;
  float* mpool = ws + o; o += (size_t)Bc * Mc * Hc;
  float* esz   = ws + o; o += (size_t)Bc * Ec * SIZE_E;
  float* msz   = ws + o; o += (size_t)Bc * Mc * SIZE_E;
  float* rctx  = ws + o; o += (size_t)Bc * Rc * Hc;
  float* cctx  = ws + o; o += (size_t)Bc * Rc * Hc;
  float* nrep  = ws + o; o += (size_t)Bc * Ec * WS;
  float* erep  = ws + o; o += (size_t)Bc * Mc * WS;
  float* rrep  = ws + o; o += (size_t)Bc * Rc * WR;
  float* crep  = ws + o; o += (size_t)Bc * Rc * WR;
  float* h1    = ws + o; o += (size_t)256 * Hc;
  float* h2    = ws + o; o += (size_t)256 * Hc;

  auto gemm = [&](const float* A, const float* W, const float* bias, float* C,
                  int Mr, int Nc, int Kd) {
    int tiles = (Mr / 16) * (Nc / 16);
    int blocks = (tiles + 7) / 8;
    wmma_gemm_relu<<<blocks, 256, 0, stream>>>(A, W, bias, C, Mr, Nc, Kd);
  };
  auto head = [&](const float* A, const float* W, const float* bias, float* C,
                  int Mr, int Nc) {
    int total = Mr * Nc;
    head_kernel<<<(total + 255) / 256, 256, 0, stream>>>(A, W, bias, C, Mr, Nc, Hc);
  };

  // --- shared front-end ---------------------------------------------------
  ctx_kernel<<<Bc, 256, 0, stream>>>(input_ids, emb, ctx);

  { int t = Bc * Ec * SIZE_E;
    sz_kernel<<<(t + 255) / 256, 256, 0, stream>>>(ent_sizes, ner_tab, esz, t); }
  { int t = Bc * Mc * SIZE_E;
    sz_kernel<<<(t + 255) / 256, 256, 0, stream>>>(men_sizes, emd_tab, msz, t); }

  pool_kernel<false><<<Bc * (Ec / 8), 256, 0, stream>>>(ent_masks, emb, epool, Ec);
  pool_kernel<false><<<Bc * (Mc / 8), 256, 0, stream>>>(men_masks, emb, mpool, Mc);
  pool_kernel<true ><<<Bc * (Rc / 8), 256, 0, stream>>>(rel_masks, emb, rctx, Rc);
  pool_kernel<true ><<<Bc * (Rc / 8), 256, 0, stream>>>(ref_masks, emb, cctx, Rc);

  // --- entity (NER) branch ------------------------------------------------
  { int t = Bc * Ec * WS;
    span_rep_kernel<<<(t + 255) / 256, 256, 0, stream>>>(ctx, epool, esz, nrep, Ec, t); }
  gemm(nrep, ner_rep_w, ner_rep_b, h1, Bc * Ec, Hc, WS);
  head(h1, ner_hd_w, ner_hd_b, out + 0, Bc * Ec, 10);              // entity_clf [2,64,10]

  // --- relation branch ----------------------------------------------------
  { int t = Bc * Rc * WR;
    rel_rep_kernel<<<(t + 255) / 256, 256, 0, stream>>>(rctx, epool, esz, relations,
                                                        rrep, Rc, Ec, t); }
  gemm(rrep, rel_w1, rel_b1, h1, Bc * Rc, Hc, WR);
  gemm(h1,   rel_w2, rel_b2, h2, Bc * Rc, Hc, Hc);
  head(h2, rel_w3, rel_b3, out + 1280, Bc * Rc, 8);                // rel_clf [2,128,8]

  // --- mention (EMD) branch ----------------------------------------------
  { int t = Bc * Mc * WS;
    span_rep_kernel<<<(t + 255) / 256, 256, 0, stream>>>(ctx, mpool, msz, erep, Mc, t); }
  gemm(erep, emd_rep_w, emd_rep_b, h1, Bc * Mc, Hc, WS);
  head(h1, emd_hd_w, emd_hd_b, out + 3328, Bc * Mc, 2);            // mention_clf [2,64,2]

  // --- coreference branch -------------------------------------------------
  { int t = Bc * Rc * WR;
    rel_rep_kernel<<<(t + 255) / 256, 256, 0, stream>>>(cctx, mpool, msz, references,
                                                        crep, Rc, Mc, t); }
  gemm(crep, cr_w1, cr_b1, h1, Bc * Rc, Hc, WR);
  gemm(h1,   cr_w2, cr_b2, h2, Bc * Rc, Hc, Hc);
  head(h2, cr_w3, cr_b3, out + 3584, Bc * Rc, 1);                  // ref_clf [2,128,1]
}